// cubic_attention_6408091206089
// MI455X (gfx1250) — compile-verified
//
#include <hip/hip_runtime.h>
#include <hip/hip_bf16.h>
#include <math.h>

// Problem constants (from reference): N=4, C=128, H=W=256, G=4, K=5, pad=2.
#define NB   4
#define CC   128
#define HH   256
#define WW   256
#define GK   20   // G*K = 4*5

typedef __attribute__((ext_vector_type(2))) float v2f;
typedef __attribute__((ext_vector_type(8))) float v8f;

// ---------------------------------------------------------------------------
// K1: per-(n,c) plane: row means (gapx_h, mean over W) and plane mean (meanx).
// One block per (n,c). 8 waves; each wave owns rows h = wid, wid+8, ...
// Lane-coalesced reads; wave32 xor-shuffle reduction.
// ---------------------------------------------------------------------------
__global__ void k_rowmean(const float* __restrict__ x,
                          float* __restrict__ gapx_h,   // [N*C*H] row means
                          float* __restrict__ meanx) {  // [N*C]   plane means
  const int b = blockIdx.x;                       // n*C + c  (512 blocks)
  const size_t base = (size_t)b * (HH * WW);
  const int lane = threadIdx.x & 31;
  const int wid  = threadIdx.x >> 5;              // 0..7
  __shared__ float wsum[8];
  float tot = 0.f;
  for (int h = wid; h < HH; h += 8) {
    const float* row = x + base + (size_t)h * WW;
    float s = 0.f;
#pragma unroll
    for (int j = 0; j < WW / 32; ++j) s += row[lane + 32 * j];
#pragma unroll
    for (int off = 16; off > 0; off >>= 1) s += __shfl_xor(s, off, 32);
    if (lane == 0) gapx_h[b * HH + h] = s * (1.f / WW);
    tot += s;                                     // full row sum on every lane
  }
  if (lane == 0) wsum[wid] = tot;
  __syncthreads();
  if (threadIdx.x == 0) {
    float t = 0.f;
    for (int i = 0; i < 8; ++i) t += wsum[i];
    meanx[b] = t * (1.f / (HH * WW));
  }
}

// ---------------------------------------------------------------------------
// K2: f = tanh(mean(4x128) @ Wm(20x128)^T) via V_WMMA_F32_16X16X4_F32.
// Single wave32. Two 16-wide N-tiles cover N=20; 32 K-steps of 4.
// A (16x4 f32) layout: lanes 0-15 -> {K=k0,k0+1}, lanes 16-31 -> {K=k0+2,k0+3}
// B (4x16 f32) layout: symmetric (row K striped across lanes within VGPR).
// Rows M>=4 / cols N>=20 are clamped loads (finite); their D entries are
// never stored, and matmul rows/cols are independent, so results are exact.
// ---------------------------------------------------------------------------
__global__ void k_wmma_f(const float* __restrict__ mean,  // [4 x 128]
                         const float* __restrict__ Wm,    // [20 x 128]
                         float* __restrict__ f) {         // [4 x 20]
  const int lane = threadIdx.x;       // 0..31, all active (EXEC all ones)
  const int half = lane >> 4;
  const int mr   = lane & 15;
  const int arow = mr < (NB - 1) ? mr : (NB - 1);      // clamp A row
  const int b0   = mr;                                 // tile0 col (0..15)
  const int b1   = (16 + mr) < GK ? (16 + mr) : (GK - 1);
  v8f acc0 = {};
  v8f acc1 = {};
  for (int k0 = 0; k0 < CC; k0 += 4) {
    const int klo = k0 + (half ? 2 : 0);
    v2f a;  a.x  = mean[arow * CC + klo];  a.y  = mean[arow * CC + klo + 1];
    v2f bA; bA.x = Wm[b0 * CC + klo];      bA.y = Wm[b0 * CC + klo + 1];
    v2f bB; bB.x = Wm[b1 * CC + klo];      bB.y = Wm[b1 * CC + klo + 1];
    acc0 = __builtin_amdgcn_wmma_f32_16x16x4_f32(false, a, false, bA,
                                                 (short)0, acc0, false, false);
    acc1 = __builtin_amdgcn_wmma_f32_16x16x4_f32(false, a, false, bB,
                                                 (short)0, acc1, false, false);
  }
  // D layout: VGPR r, lanes 0-15 hold D[M=r][N=lane].
  if (lane < 16) {
#pragma unroll
    for (int r = 0; r < NB; ++r) {
      f[r * GK + lane] = tanhf(acc0[r]);
      const int col = 16 + lane;
      if (col < GK) f[r * GK + col] = tanhf(acc1[r]);
    }
  }
}

__device__ __forceinline__ int reflect256(int i) {
  i = i < 0 ? -i : i;
  return i > 255 ? 510 - i : i;
}

// ---------------------------------------------------------------------------
// K3: horizontal strip pass. One block per (n,c,h) row; row staged in LDS,
// 5-tap reflect conv along W, blended with gapx_h and per-channel scalars.
// ---------------------------------------------------------------------------
__global__ void k_hpass(const float* __restrict__ x, const float* __restrict__ f,
                        const float* __restrict__ gapx_h,
                        const float* __restrict__ inside,
                        const float* __restrict__ lamb_l,
                        const float* __restrict__ lamb_h,
                        float* __restrict__ tmp) {
  const int r  = blockIdx.x;          // (n*C + c)*H + h
  const int nc = r >> 8;              // n*C + c   (H == 256)
  const int c  = nc & (CC - 1);
  const int n  = nc >> 7;             // C == 128
  const int g  = c >> 5;              // c / (C/G), C/G == 32
  const size_t base = (size_t)r * WW;
  const int t = threadIdx.x;
  __shared__ float row[WW];
  row[t] = x[base + t];
  __syncthreads();
  const float* fr = f + n * GK + g * 5;
  float conv = 0.f;
#pragma unroll
  for (int k = 0; k < 5; ++k) conv += fr[k] * row[reflect256(t + k - 2)];
  const float ins = inside[c];
  const float o = lamb_l[c] * ((ins + 1.f) * conv - ins * gapx_h[r]) +
                  (lamb_h[c] + 1.f) * row[t];
  tmp[base + t] = o;
}

// ---------------------------------------------------------------------------
// K4: per-(n,c): column means over H of tmp (gapx_v) and plane mean (mean1).
// Thread t owns column w=t; coalesced across lanes each h-iteration.
// ---------------------------------------------------------------------------
__global__ void k_colmean(const float* __restrict__ tmp,
                          float* __restrict__ gapx_v,  // [N*C*W]
                          float* __restrict__ mean1) { // [N*C]
  const int b = blockIdx.x;                      // n*C + c
  const size_t base = (size_t)b * (HH * WW);
  const int t = threadIdx.x;                     // w
  float s = 0.f;
  for (int h = 0; h < HH; ++h) s += tmp[base + (size_t)h * WW + t];
  gapx_v[b * WW + t] = s * (1.f / HH);
#pragma unroll
  for (int off = 16; off > 0; off >>= 1) s += __shfl_xor(s, off, 32);
  __shared__ float wsum[8];
  if ((t & 31) == 0) wsum[t >> 5] = s;
  __syncthreads();
  if (t == 0) {
    float tt = 0.f;
    for (int i = 0; i < 8; ++i) tt += wsum[i];
    mean1[b] = tt * (1.f / (HH * WW));
  }
}

// ---------------------------------------------------------------------------
// K6: vertical strip pass + final affine combine with original x.
// One block per (n,c,h) row; 5 coalesced strided row-reads of tmp (the whole
// 128 MB intermediate fits in the 192 MB L2, so the 5x re-read is cheap).
// ---------------------------------------------------------------------------
__global__ void k_vpass_final(const float* __restrict__ tmp,
                              const float* __restrict__ x,
                              const float* __restrict__ f,
                              const float* __restrict__ gapx_v,
                              const float* __restrict__ inside,
                              const float* __restrict__ lamb_l,
                              const float* __restrict__ lamb_h,
                              const float* __restrict__ gamma,
                              const float* __restrict__ beta,
                              float* __restrict__ out) {
  const int r  = blockIdx.x;          // (n*C + c)*H + h
  const int h  = r & (HH - 1);
  const int nc = r >> 8;
  const int c  = nc & (CC - 1);
  const int n  = nc >> 7;
  const int g  = c >> 5;
  const int t  = threadIdx.x;         // w
  const size_t pb = (size_t)nc * (HH * WW);
  const float* fr = f + n * GK + g * 5;
  float conv = 0.f, center = 0.f;
#pragma unroll
  for (int k = 0; k < 5; ++k) {
    const int hh = reflect256(h + k - 2);
    const float v = tmp[pb + (size_t)hh * WW + t];
    if (k == 2) center = v;           // reflect(h) == h for in-range h
    conv += fr[k] * v;
  }
  const float ins = inside[c];
  const float o2 = lamb_l[c] * ((ins + 1.f) * conv - ins * gapx_v[nc * WW + t]) +
                   (lamb_h[c] + 1.f) * center;
  const size_t base = (size_t)r * WW;
  out[base + t] = gamma[c] * o2 + beta[c] * x[base + t];
}

// ---------------------------------------------------------------------------
// Launch. Workspace layout (floats):
//   [0, 33554432)                 tmp   (intermediate after horizontal pass)
//   [+0, +512)                    meanx
//   [+512, +1024)                 mean1
//   [+1024, +1152)                fh (80 used)
//   [+1152, +1280)                fv (80 used)
//   [+1280, +132352)              gapx_h (N*C*H = 131072)
//   [+132352, +263424)            gapx_v (N*C*W = 131072)
// Total ~135 MB.
// ---------------------------------------------------------------------------
extern "C" void kernel_launch(void* const* d_in, const int* in_sizes, int n_in,
                              void* d_out, int out_size, void* d_ws, size_t ws_size,
                              hipStream_t stream) {
  const float* x        = (const float*)d_in[0];
  const float* conv_h   = (const float*)d_in[1];
  const float* inside_h = (const float*)d_in[2];
  const float* lamb_l_h = (const float*)d_in[3];
  const float* lamb_h_h = (const float*)d_in[4];
  const float* conv_v   = (const float*)d_in[5];
  const float* inside_v = (const float*)d_in[6];
  const float* lamb_l_v = (const float*)d_in[7];
  const float* lamb_h_v = (const float*)d_in[8];
  const float* gamma    = (const float*)d_in[9];
  const float* beta     = (const float*)d_in[10];

  float* ws    = (float*)d_ws;
  float* tmp   = ws;
  float* meanx = ws + (size_t)NB * CC * HH * WW;   // +33554432
  float* mean1 = meanx + 512;
  float* fh    = mean1 + 512;
  float* fv    = fh + 128;
  float* gh    = fv + 128;                          // gapx_h
  float* gv    = gh + (size_t)NB * CC * HH;         // gapx_v
  float* out   = (float*)d_out;

  const int planes = NB * CC;          // 512
  const int rows   = NB * CC * HH;     // 131072

  k_rowmean<<<planes, 256, 0, stream>>>(x, gh, meanx);
  k_wmma_f<<<1, 32, 0, stream>>>(meanx, conv_h, fh);
  k_hpass<<<rows, WW, 0, stream>>>(x, fh, gh, inside_h, lamb_l_h, lamb_h_h, tmp);
  k_colmean<<<planes, WW, 0, stream>>>(tmp, gv, mean1);
  k_wmma_f<<<1, 32, 0, stream>>>(mean1, conv_v, fv);
  k_vpass_final<<<rows, WW, 0, stream>>>(tmp, x, fv, gv, inside_v, lamb_l_v,
                                         lamb_h_v, gamma, beta, out);
}